// MaceLayer_36146444763353
// MI455X (gfx1250) — compile-verified
//
#include <hip/hip_runtime.h>

typedef __attribute__((ext_vector_type(2))) float v2f;
typedef __attribute__((ext_vector_type(8))) float v8f;

#define NN 50000
#define NE 400000
#define CC 64
#define NS 10
#define NR 8

__device__ __forceinline__ v8f wmma4(v2f a, v2f b, v8f c) {
  // D = A(16x4,f32) * B(4x16,f32) + C(16x16,f32)
  return __builtin_amdgcn_wmma_f32_16x16x4_f32(false, a, false, b, (short)0, c,
                                               false, false);
}

__device__ __forceinline__ float silu_f(float x) {
  // v_exp_f32 + v_rcp_f32 (both TRANS pipe, co-issue with VALU).
  // Denominator is in [1, inf): rcp is ~1 ulp here, numerically safe.
  return x * __builtin_amdgcn_rcpf(1.0f + __expf(-x));
}

__global__ void zero_kernel(float4* __restrict__ p, int n4) {
  int i = blockIdx.x * blockDim.x + threadIdx.x;
  int stride = gridDim.x * blockDim.x;
  const float4 z = make_float4(0.f, 0.f, 0.f, 0.f);
  for (; i < n4; i += stride) p[i] = z;
}

// One wave (32 threads) per block, 16 edges per block.
__global__ void edge_kernel(const float* __restrict__ edge_vectors,
                            const float* __restrict__ node_feats,
                            const float* __restrict__ radial,
                            const int* __restrict__ senders,
                            const int* __restrict__ receivers,
                            const float* __restrict__ W_r1,
                            const float* __restrict__ W_r2,
                            const float* __restrict__ W_r3,
                            float* __restrict__ a0,
                            float* __restrict__ a1) {
  __shared__ __align__(16) float lds_h1[16 * 64];
  __shared__ __align__(16) float lds_h2[16 * 64];
  __shared__ __align__(16) float lds_w[4 * 16 * 16];  // [j][edge][chan]

  const int lane = threadIdx.x & 31;
  const int lrow = lane & 15;   // M (edge) for A-frags / N (col) for B-frags
  const int lhalf = lane >> 4;  // lane half
  const int ebase = blockIdx.x * 16;

  // ---------- GEMM1: H1 = silu(RE[16x8] @ W_r1[8x64]) ----------
  v2f aK0, aK1;
  {
    const float* rr = radial + (size_t)(ebase + lrow) * NR;
    aK0.x = rr[0 + 2 * lhalf];
    aK0.y = rr[1 + 2 * lhalf];
    aK1.x = rr[4 + 2 * lhalf];
    aK1.y = rr[5 + 2 * lhalf];
  }
#pragma unroll
  for (int nt = 0; nt < 4; ++nt) {
    const int ncol = nt * 16 + lrow;
    v2f b0, b1;
    b0.x = W_r1[(0 + 2 * lhalf) * 64 + ncol];
    b0.y = W_r1[(1 + 2 * lhalf) * 64 + ncol];
    b1.x = W_r1[(4 + 2 * lhalf) * 64 + ncol];
    b1.y = W_r1[(5 + 2 * lhalf) * 64 + ncol];
    v8f acc = {};
    acc = wmma4(aK0, b0, acc);
    acc = wmma4(aK1, b1, acc);
#pragma unroll
    for (int g = 0; g < 8; ++g)
      lds_h1[(g + 8 * lhalf) * 64 + ncol] = silu_f(acc[g]);
  }
  __syncthreads();

  // ---------- GEMM2: H2 = silu(H1[16x64] @ W_r2[64x64]) ----------
#pragma unroll
  for (int nt = 0; nt < 4; ++nt) {
    const int ncol = nt * 16 + lrow;
    v8f acc = {};
    for (int kk = 0; kk < 16; ++kk) {
      const int k0 = kk * 4 + 2 * lhalf;
      v2f a, b;
      a.x = lds_h1[lrow * 64 + k0];
      a.y = lds_h1[lrow * 64 + k0 + 1];
      b.x = W_r2[k0 * 64 + ncol];
      b.y = W_r2[(k0 + 1) * 64 + ncol];
      acc = wmma4(a, b, acc);
    }
#pragma unroll
    for (int g = 0; g < 8; ++g)
      lds_h2[(g + 8 * lhalf) * 64 + ncol] = silu_f(acc[g]);
  }
  __syncthreads();

  // Per-edge geometry (each lane-half handles the same 16 edges).
  const int e = ebase + lrow;
  const int snd = senders[e];
  const int rcv = receivers[e];
  const float ex = edge_vectors[(size_t)e * 3 + 0];
  const float ey = edge_vectors[(size_t)e * 3 + 1];
  const float ez = edge_vectors[(size_t)e * 3 + 2];
  const float rinv = 1.0f / sqrtf(ex * ex + ey * ey + ez * ez + 1e-12f);
  const float y0 = 1.7320508075688772f * ex * rinv;
  const float y1 = 1.7320508075688772f * ey * rinv;
  const float y2 = 1.7320508075688772f * ez * rinv;

  // ---------- GEMM3 (W = H2 @ W_r3[64x256]) fused with message scatter ----
  for (int cc = 0; cc < 4; ++cc) {  // 16-channel chunk
#pragma unroll
    for (int j = 0; j < 4; ++j) {   // w[:, j, :] component
      const int nt = j * 4 + cc;
      const int ncol = nt * 16 + lrow;
      v8f acc = {};
      for (int kk = 0; kk < 16; ++kk) {
        const int k0 = kk * 4 + 2 * lhalf;
        v2f a, b;
        a.x = lds_h2[lrow * 64 + k0];
        a.y = lds_h2[lrow * 64 + k0 + 1];
        b.x = W_r3[k0 * 256 + ncol];
        b.y = W_r3[(k0 + 1) * 256 + ncol];
        acc = wmma4(a, b, acc);
      }
#pragma unroll
      for (int g = 0; g < 8; ++g)
        lds_w[(j * 16 + (g + 8 * lhalf)) * 16 + lrow] = acc[g];
    }
    __syncthreads();

    // Scatter: lane handles edge lrow, 8 channels of this 16-chunk.
#pragma unroll
    for (int q = 0; q < 8; ++q) {
      const int cl = lhalf * 8 + q;
      const int c = cc * 16 + cl;
      const float w0 = lds_w[(0 * 16 + lrow) * 16 + cl];
      const float w1 = lds_w[(1 * 16 + lrow) * 16 + cl];
      const float w2 = lds_w[(2 * 16 + lrow) * 16 + cl];
      const float w3 = lds_w[(3 * 16 + lrow) * 16 + cl];
      const float4 xs = *reinterpret_cast<const float4*>(
          node_feats + (size_t)snd * 256 + (size_t)c * 4);
      const float dv = xs.y * y0 + xs.z * y1 + xs.w * y2;
      atomicAdd(&a0[(size_t)rcv * 64 + c], w0 * xs.x + w1 * dv);
      const float t = w2 * xs.x;
      const size_t bi = ((size_t)rcv * 64 + c) * 3;
      atomicAdd(&a1[bi + 0], t * y0 + w3 * xs.y);
      atomicAdd(&a1[bi + 1], t * y1 + w3 * xs.z);
      atomicAdd(&a1[bi + 2], t * y2 + w3 * xs.w);
    }
    __syncthreads();
  }
}

// One wave per block, 16 nodes per block.
__global__ void node_kernel(const float* __restrict__ node_feats,
                            const int* __restrict__ species,
                            const float* __restrict__ a0,
                            const float* __restrict__ a1,
                            const float* __restrict__ W_lin0,
                            const float* __restrict__ W_lin1,
                            const float* __restrict__ Wsc0,
                            const float* __restrict__ Wsc1,
                            const float* __restrict__ Wp0,
                            const float* __restrict__ Wp1,
                            const float* __restrict__ Wres0,
                            const float* __restrict__ Wres1,
                            const float* __restrict__ W_read,
                            float* __restrict__ out0,
                            float* __restrict__ out_feats) {
  __shared__ __align__(16) float lds_x[16 * 256];      // (n, c, 4)
  __shared__ __align__(16) float lds_B0[16 * 64];
  __shared__ __align__(16) float lds_B1[16 * 64 * 3];
  __shared__ float lds_red[16];

  const int lane = threadIdx.x & 31;
  const int lrow = lane & 15;
  const int lhalf = lane >> 4;
  const int nb = blockIdx.x * 16;

  // Stage node_feats tile (16 nodes x 256 floats) into LDS.
  for (int t = lane; t < 16 * 64; t += 32)
    reinterpret_cast<float4*>(lds_x)[t] =
        reinterpret_cast<const float4*>(node_feats + (size_t)nb * 256)[t];
  if (lane < 16) lds_red[lane] = 0.0f;
  __syncthreads();

  int spc[8];
#pragma unroll
  for (int g = 0; g < 8; ++g) spc[g] = species[nb + g + 8 * lhalf];

  // ---------- A0 = (a0/8) @ W_lin0 ; A1_i = (a1_i/8) @ W_lin1 ----------
  // then elementwise species polynomial -> B0, B1 into LDS.
  const float* a0row = a0 + (size_t)(nb + lrow) * 64;
  const float* a1row = a1 + (size_t)(nb + lrow) * 64 * 3;
  for (int nt = 0; nt < 4; ++nt) {
    const int ncol = nt * 16 + lrow;
    v8f A0acc = {};
    v8f A1acc[3] = {{}, {}, {}};
    for (int kk = 0; kk < 16; ++kk) {
      const int k0 = kk * 4 + 2 * lhalf;
      v2f b0, b1;
      b0.x = W_lin0[k0 * 64 + ncol];
      b0.y = W_lin0[(k0 + 1) * 64 + ncol];
      b1.x = W_lin1[k0 * 64 + ncol];
      b1.y = W_lin1[(k0 + 1) * 64 + ncol];
      v2f a;
      a.x = a0row[k0] * 0.125f;
      a.y = a0row[k0 + 1] * 0.125f;
      A0acc = wmma4(a, b0, A0acc);
#pragma unroll
      for (int i = 0; i < 3; ++i) {
        v2f av;
        av.x = a1row[k0 * 3 + i] * 0.125f;
        av.y = a1row[(k0 + 1) * 3 + i] * 0.125f;
        A1acc[i] = wmma4(av, b1, A1acc[i]);
      }
    }
#pragma unroll
    for (int g = 0; g < 8; ++g) {
      const int n = g + 8 * lhalf;
      const int sp = spc[g];
      const float A0v = A0acc[g];
      const float v0 = A1acc[0][g], v1 = A1acc[1][g], v2 = A1acc[2][g];
      const float n1 = v0 * v0 + v1 * v1 + v2 * v2;
      const float* w0p = Wsc0 + ((size_t)sp * 64 + ncol) * 5;
      const float* w1p = Wsc1 + ((size_t)sp * 64 + ncol) * 4;
      const float A02 = A0v * A0v;
      const float B0 = w0p[0] * A0v + w0p[1] * A02 + w0p[2] * A02 * A0v +
                       w0p[3] * n1 + w0p[4] * A0v * n1;
      const float s1 = w1p[0] + w1p[1] * A0v + w1p[2] * A02 + w1p[3] * n1;
      lds_B0[n * 64 + ncol] = B0;
      lds_B1[(n * 64 + ncol) * 3 + 0] = s1 * v0;
      lds_B1[(n * 64 + ncol) * 3 + 1] = s1 * v1;
      lds_B1[(n * 64 + ncol) * 3 + 2] = s1 * v2;
    }
  }
  __syncthreads();

  // ---------- res (species-specific, VALU) + F = B @ Wp + res (WMMA) -----
  for (int nt = 0; nt < 4; ++nt) {
    const int dcol = nt * 16 + lrow;
    v8f F0acc = {};
    v8f F1acc[3] = {{}, {}, {}};
#pragma unroll
    for (int g = 0; g < 8; ++g) {
      const int n = g + 8 * lhalf;
      const int sp = spc[g];
      const float* w0b = Wres0 + (size_t)sp * 4096 + dcol;
      const float* w1b = Wres1 + (size_t)sp * 4096 + dcol;
      const float* xr = lds_x + n * 256;
      float r0 = 0.f, rx = 0.f, ry = 0.f, rz = 0.f;
      for (int ci = 0; ci < 64; ++ci) {
        const float wr0 = w0b[ci * 64];
        const float wr1 = w1b[ci * 64];
        r0 += wr0 * xr[ci * 4 + 0];
        rx += wr1 * xr[ci * 4 + 1];
        ry += wr1 * xr[ci * 4 + 2];
        rz += wr1 * xr[ci * 4 + 3];
      }
      F0acc[g] = r0;
      F1acc[0][g] = rx;
      F1acc[1][g] = ry;
      F1acc[2][g] = rz;
    }
    for (int kk = 0; kk < 16; ++kk) {
      const int k0 = kk * 4 + 2 * lhalf;
      v2f bp0, bp1;
      bp0.x = Wp0[k0 * 64 + dcol];
      bp0.y = Wp0[(k0 + 1) * 64 + dcol];
      bp1.x = Wp1[k0 * 64 + dcol];
      bp1.y = Wp1[(k0 + 1) * 64 + dcol];
      v2f aB0;
      aB0.x = lds_B0[lrow * 64 + k0];
      aB0.y = lds_B0[lrow * 64 + k0 + 1];
      F0acc = wmma4(aB0, bp0, F0acc);
#pragma unroll
      for (int i = 0; i < 3; ++i) {
        v2f aB1;
        aB1.x = lds_B1[(lrow * 64 + k0) * 3 + i];
        aB1.y = lds_B1[(lrow * 64 + k0 + 1) * 3 + i];
        F1acc[i] = wmma4(aB1, bp1, F1acc[i]);
      }
    }
    // Outputs: node_feats_out (N,C,4) as float4, readout partial sums.
    const float wr = W_read[dcol];
#pragma unroll
    for (int g = 0; g < 8; ++g) {
      const int n = g + 8 * lhalf;
      float4 o;
      o.x = F0acc[g];
      o.y = F1acc[0][g];
      o.z = F1acc[1][g];
      o.w = F1acc[2][g];
      *reinterpret_cast<float4*>(out_feats +
                                 ((size_t)(nb + n) * 64 + dcol) * 4) = o;
      atomicAdd(&lds_red[n], F0acc[g] * wr);
    }
  }
  __syncthreads();
  if (lane < 16) out0[nb + lane] = lds_red[lane];
}

extern "C" void kernel_launch(void* const* d_in, const int* in_sizes, int n_in,
                              void* d_out, int out_size, void* d_ws,
                              size_t ws_size, hipStream_t stream) {
  const float* edge_vectors = (const float*)d_in[0];
  const float* node_feats = (const float*)d_in[1];
  const int* node_species = (const int*)d_in[2];
  const float* radial = (const float*)d_in[3];
  const int* senders = (const int*)d_in[4];
  const int* receivers = (const int*)d_in[5];
  const float* W_r1 = (const float*)d_in[6];
  const float* W_r2 = (const float*)d_in[7];
  const float* W_r3 = (const float*)d_in[8];
  const float* W_lin0 = (const float*)d_in[9];
  const float* W_lin1 = (const float*)d_in[10];
  const float* Wsc0 = (const float*)d_in[11];
  const float* Wsc1 = (const float*)d_in[12];
  const float* Wp0 = (const float*)d_in[13];
  const float* Wp1 = (const float*)d_in[14];
  const float* Wres0 = (const float*)d_in[15];
  const float* Wres1 = (const float*)d_in[16];
  const float* W_read = (const float*)d_in[17];

  float* a0 = (float*)d_ws;                    // N*C floats
  float* a1 = a0 + (size_t)NN * CC;            // N*C*3 floats
  float* out0 = (float*)d_out;                 // N floats
  float* out_feats = out0 + NN;                // N*C*4 floats

  zero_kernel<<<2048, 256, 0, stream>>>((float4*)d_ws, NN * CC);  // N*C*4 / 4
  edge_kernel<<<NE / 16, 32, 0, stream>>>(edge_vectors, node_feats, radial,
                                          senders, receivers, W_r1, W_r2, W_r3,
                                          a0, a1);
  node_kernel<<<NN / 16, 32, 0, stream>>>(node_feats, node_species, a0, a1,
                                          W_lin0, W_lin1, Wsc0, Wsc1, Wp0, Wp1,
                                          Wres0, Wres1, W_read, out0,
                                          out_feats);
}